// rollout_GNN_GRU_43379169689800
// MI455X (gfx1250) — compile-verified
//
#include <hip/hip_runtime.h>

// rollout GNN+GRU, MI455X / gfx1250, wave32, f32 WMMA 16x16x4.
// 3 fused kernels per step (latency-bound rollout: minimize graph nodes and
// L2 round-trips; all intermediates are L2-resident on a 192MB L2).

constexpr int N_NODES = 20000;
constexpr int IN_D    = 32;
constexpr int HID_D   = 128;
constexpr int T_STEPS = 16;
constexpr int DEG     = 16;
constexpr int LDH     = 132;   // padded LDS row stride (floats): 132 mod 64 = 4
                               // -> 16 consecutive rows hit 16 distinct bank groups

typedef float v2f __attribute__((ext_vector_type(2)));
typedef float v8f __attribute__((ext_vector_type(8)));

static __device__ inline v8f v8zero() {
    v8f v = {0.f, 0.f, 0.f, 0.f, 0.f, 0.f, 0.f, 0.f};
    return v;
}

// D = A(16x4) * B(4x16) + C via V_WMMA_F32_16X16X4_F32 (f32 matrix pipe:
// reference precision; 56 GFLOP total is nowhere near any matrix ceiling).
static __device__ inline v8f wmma_f32(v2f a, v2f b, v8f c) {
    return __builtin_amdgcn_wmma_f32_16x16x4_f32(
        false, a, false, b, (short)0, c, false, false);
}

// 16x4 operand tile from row-major global M[ld] (B operands: W^T column j == W row j).
static __device__ inline v2f load_tile(const float* __restrict__ M, int ld,
                                       int rowBase, int kk, int lane) {
    int row = rowBase + (lane & 15);
    int kb  = kk + ((lane >> 4) << 1);
    const float* p = M + (size_t)row * ld + kb;
    v2f v; v.x = p[0]; v.y = p[1];
    return v;
}

// 16x4 operand tile from LDS (padded stride LDH).
static __device__ inline v2f load_tile_lds(const float* L, int rowBase, int kk, int lane) {
    int row = rowBase + (lane & 15);
    int kb  = kk + ((lane >> 4) << 1);
    const float* p = L + row * LDH + kb;
    v2f v; v.x = p[0]; v.y = p[1];
    return v;
}

static __device__ inline float sigmoidf(float x) { return 1.0f / (1.0f + __expf(-x)); }

// ---------------------------------------------------------------------------
// init: state <- x0, hidden <- 0
// ---------------------------------------------------------------------------
__global__ void __launch_bounds__(256) init_kernel(const float* __restrict__ x0,
                                                   float* __restrict__ state,
                                                   float* __restrict__ h) {
    int idx = blockIdx.x * 256 + threadIdx.x;
    if (idx < N_NODES * HID_D) h[idx] = 0.f;
    if (idx < N_NODES * IN_D)  state[idx] = x0[idx];
}

// ---------------------------------------------------------------------------
// K1: GRU -> (LDS) -> z1 = h' @ fc1^T -> attention dots s1,s2.
// Block = 8 waves = 2 Mtiles x 4 column chunks = 32 nodes, all 128 columns.
// grid 625 x 256 (exact; EXEC all-ones at every WMMA).
// ---------------------------------------------------------------------------
__global__ void __launch_bounds__(256) gru_z1_kernel(
    const float* __restrict__ x,    const float* __restrict__ h_in,
    float* __restrict__ h_out,
    const float* __restrict__ W_ih, const float* __restrict__ W_hh,
    const float* __restrict__ b_ih, const float* __restrict__ b_hh,
    const float* __restrict__ fc1,  const float* __restrict__ attn1,
    float* __restrict__ z1, float* __restrict__ s1, float* __restrict__ s2) {
    __shared__ float h_lds[32 * LDH];
    __shared__ float dotsP[32][4];
    __shared__ float dotsQ[32][4];

    const int tid  = threadIdx.x;
    const int lane = tid & 31;
    const int w    = tid >> 5;
    const int sub  = lane & 15;
    const int half = lane >> 4;
    const int mtl  = w >> 2;              // local 16-node group (0/1)
    const int c    = w & 3;               // 32-column chunk
    const int mbase = (blockIdx.x * 2 + mtl) * 16;
    const int jb0   = c * 32;

    // ---- Phase A: GRU (r/z merged input+hidden accumulators; ni/nh split) ----
    v8f aR[2], aZ[2], aNI[2], aNH[2];
#pragma unroll
    for (int s = 0; s < 2; ++s) { aR[s] = v8zero(); aZ[s] = v8zero(); aNI[s] = v8zero(); aNH[s] = v8zero(); }

    for (int kk = 0; kk < IN_D; kk += 4) {          // x @ W_ih^T
        v2f a = load_tile(x, IN_D, mbase, kk, lane);
#pragma unroll
        for (int s = 0; s < 2; ++s) {
            int j = jb0 + s * 16;
            aR[s]  = wmma_f32(a, load_tile(W_ih, IN_D, j,             kk, lane), aR[s]);
            aZ[s]  = wmma_f32(a, load_tile(W_ih, IN_D, HID_D + j,     kk, lane), aZ[s]);
            aNI[s] = wmma_f32(a, load_tile(W_ih, IN_D, 2 * HID_D + j, kk, lane), aNI[s]);
        }
    }
    for (int kk = 0; kk < HID_D; kk += 4) {         // h @ W_hh^T
        v2f a = load_tile(h_in, HID_D, mbase, kk, lane);
#pragma unroll
        for (int s = 0; s < 2; ++s) {
            int j = jb0 + s * 16;
            aR[s]  = wmma_f32(a, load_tile(W_hh, HID_D, j,             kk, lane), aR[s]);
            aZ[s]  = wmma_f32(a, load_tile(W_hh, HID_D, HID_D + j,     kk, lane), aZ[s]);
            aNH[s] = wmma_f32(a, load_tile(W_hh, HID_D, 2 * HID_D + j, kk, lane), aNH[s]);
        }
    }
#pragma unroll
    for (int s = 0; s < 2; ++s) {
        int col   = jb0 + s * 16 + sub;             // column uniform across the 8 acc elems
        float br  = b_ih[col] + b_hh[col];
        float bz  = b_ih[HID_D + col] + b_hh[HID_D + col];
        float bni = b_ih[2 * HID_D + col];
        float bnh = b_hh[2 * HID_D + col];
#pragma unroll
        for (int i = 0; i < 8; ++i) {
            int m    = i + (half << 3);
            int node = mbase + m;
            float r  = sigmoidf(aR[s][i] + br);
            float zg = sigmoidf(aZ[s][i] + bz);
            float nc = tanhf(aNI[s][i] + bni + r * (aNH[s][i] + bnh));
            float ho = h_in[(size_t)node * HID_D + col];
            float hn = (1.f - zg) * nc + zg * ho;
            h_out[(size_t)node * HID_D + col] = hn;             // for next step's GRU
            h_lds[(mtl * 16 + m) * LDH + col] = hn;             // for fused z1 GEMM
        }
    }
    __syncthreads();

    // ---- Phase B: z1[16x32 chunk] = h'(LDS) @ fc1^T ----
    v8f zacc[2];
#pragma unroll
    for (int s = 0; s < 2; ++s) zacc[s] = v8zero();
    for (int kk = 0; kk < HID_D; kk += 4) {
        v2f a = load_tile_lds(h_lds, mtl * 16, kk, lane);
#pragma unroll
        for (int s = 0; s < 2; ++s)
            zacc[s] = wmma_f32(a, load_tile(fc1, HID_D, jb0 + s * 16, kk, lane), zacc[s]);
    }

    // ---- Phase C: store z1 + attention-dot partials (s1 = a1.z, s2 = a2.z) ----
    float pi[8] = {0, 0, 0, 0, 0, 0, 0, 0};
    float qi[8] = {0, 0, 0, 0, 0, 0, 0, 0};
#pragma unroll
    for (int s = 0; s < 2; ++s) {
        int colg = jb0 + s * 16 + sub;
        float a1 = attn1[colg];
        float a2 = attn1[HID_D + colg];
#pragma unroll
        for (int i = 0; i < 8; ++i) {
            int m     = i + (half << 3);
            int gnode = blockIdx.x * 32 + mtl * 16 + m;
            float zv  = zacc[s][i];
            z1[(size_t)gnode * HID_D + colg] = zv;
            pi[i] += a1 * zv;
            qi[i] += a2 * zv;
        }
    }
#pragma unroll
    for (int i = 0; i < 8; ++i) {
#pragma unroll
        for (int msk = 1; msk < 16; msk <<= 1) {    // 16-lane half reduction
            pi[i] += __shfl_xor(pi[i], msk, 32);
            qi[i] += __shfl_xor(qi[i], msk, 32);
        }
    }
    if (sub == 0) {
#pragma unroll
        for (int i = 0; i < 8; ++i) {
            int nl = mtl * 16 + (half << 3) + i;
            dotsP[nl][c] = pi[i];
            dotsQ[nl][c] = qi[i];
        }
    }
    __syncthreads();
    if (tid < 64) {
        int which = tid >> 5, nl = tid & 31;
        const float (*d)[4] = which ? dotsQ : dotsP;
        float v = d[nl][0] + d[nl][1] + d[nl][2] + d[nl][3];
        int g = blockIdx.x * 32 + nl;
        if (which) s2[g] = v; else s1[g] = v;
    }
}

// ---------------------------------------------------------------------------
// K2: GAT1 softmax-aggregate -> relu -> (LDS) -> z2 = g1 @ fc2^T -> dots2.
// Block = 16 nodes (half-wave per node; dst = repeat(arange) => 16 contiguous
// edges per node, segment ops become 16-lane shuffle reductions).
// grid 1250 x 256 (exact).
// ---------------------------------------------------------------------------
__global__ void __launch_bounds__(256) agg1_z2_kernel(
    const float* __restrict__ z1, const float* __restrict__ s1,
    const float* __restrict__ s2, const int* __restrict__ src,
    const int* __restrict__ dst,
    const float* __restrict__ fc2, const float* __restrict__ attn2,
    float* __restrict__ z2, float* __restrict__ s1b, float* __restrict__ s2b) {
    __shared__ float g1l[16 * LDH];
    __shared__ float z2l[16 * 33];

    const int tid   = threadIdx.x;
    const int lane  = tid & 31;
    const int w     = tid >> 5;
    const int sub   = lane & 15;
    const int half  = lane >> 4;
    const int lnode = w * 2 + half;                 // 0..15
    const int node  = blockIdx.x * 16 + lnode;

    // ---- Phase A: edge softmax + weighted gather, relu, stage to LDS ----
    const int e  = node * DEG + sub;
    const int se = src[e];
    const int de = dst[e];
    float sc = s1[se] + s2[de];
    sc = sc > 0.f ? sc : 0.01f * sc;                // leaky_relu
    float mx = sc;
#pragma unroll
    for (int m = 1; m < 16; m <<= 1) mx = fmaxf(mx, __shfl_xor(mx, m, 32));
    float ee = __expf(sc - mx);
    float dn = ee;
#pragma unroll
    for (int m = 1; m < 16; m <<= 1) dn += __shfl_xor(dn, m, 32);
    float alpha = ee / dn;

    int   sjs[16];                                  // hoisted convergent shuffles
    float ajs[16];
#pragma unroll
    for (int j = 0; j < 16; ++j) {
        sjs[j] = __shfl(se,    half * 16 + j, 32);
        ajs[j] = __shfl(alpha, half * 16 + j, 32);
    }
#pragma unroll
    for (int it = 0; it < HID_D / 16; ++it) {
        int f = sub + it * 16;
        float acc = 0.f;
#pragma unroll
        for (int j = 0; j < 16; ++j)
            acc += ajs[j] * z1[(size_t)sjs[j] * HID_D + f];
        g1l[lnode * LDH + f] = fmaxf(acc, 0.f);     // fused relu
    }
    __syncthreads();

    // ---- Phase B: z2[16x32] = relu(g1)(LDS) @ fc2^T (waves 0,1) ----
    if (w < 2) {
        v8f zacc = v8zero();
        for (int kk = 0; kk < HID_D; kk += 4) {
            v2f a = load_tile_lds(g1l, 0, kk, lane);
            zacc  = wmma_f32(a, load_tile(fc2, HID_D, w * 16, kk, lane), zacc);
        }
        int colg = w * 16 + sub;
#pragma unroll
        for (int i = 0; i < 8; ++i) {
            int m = i + (half << 3);
            int g = blockIdx.x * 16 + m;
            z2[(size_t)g * IN_D + colg] = zacc[i];
            z2l[m * 33 + colg]          = zacc[i];
        }
    }
    __syncthreads();

    // ---- Phase C: dots2 (s1b = a1.z2, s2b = a2.z2), 2 nodes per wave ----
    const float* zr = &z2l[lnode * 33];
    float p = attn2[sub] * zr[sub] + attn2[16 + sub] * zr[16 + sub];
    float q = attn2[32 + sub] * zr[sub] + attn2[48 + sub] * zr[16 + sub];
#pragma unroll
    for (int m = 1; m < 16; m <<= 1) {
        p += __shfl_xor(p, m, 32);
        q += __shfl_xor(q, m, 32);
    }
    if (sub == 0) { s1b[node] = p; s2b[node] = q; }
}

// ---------------------------------------------------------------------------
// K3: GAT2 softmax-aggregate + Euler update + output write.
// out[t][n][:32] = state (pre-update), out[t][n][32:] = xii, state += dt*xii.
// grid 1250 x 256 (exact).
// ---------------------------------------------------------------------------
__global__ void __launch_bounds__(256) agg2_final_kernel(
    const float* __restrict__ z2, const float* __restrict__ s1b,
    const float* __restrict__ s2b, const int* __restrict__ src,
    const int* __restrict__ dst,
    float* __restrict__ state, float* __restrict__ out_t,
    const float* __restrict__ tarr, int step) {
    const int lane = threadIdx.x & 31;
    const int sub  = lane & 15;
    const int half = lane >> 4;
    const int node = blockIdx.x * 16 + (threadIdx.x >> 5) * 2 + half;

    const int e  = node * DEG + sub;
    const int se = src[e];
    const int de = dst[e];
    float sc = s1b[se] + s2b[de];
    sc = sc > 0.f ? sc : 0.01f * sc;
    float mx = sc;
#pragma unroll
    for (int m = 1; m < 16; m <<= 1) mx = fmaxf(mx, __shfl_xor(mx, m, 32));
    float ee = __expf(sc - mx);
    float dn = ee;
#pragma unroll
    for (int m = 1; m < 16; m <<= 1) dn += __shfl_xor(dn, m, 32);
    float alpha = ee / dn;

    int   sjs[16];
    float ajs[16];
#pragma unroll
    for (int j = 0; j < 16; ++j) {
        sjs[j] = __shfl(se,    half * 16 + j, 32);
        ajs[j] = __shfl(alpha, half * 16 + j, 32);
    }
    const float dt = (step < T_STEPS - 1) ? (tarr[step + 1] - tarr[step]) : 0.f;
#pragma unroll
    for (int it = 0; it < IN_D / 16; ++it) {
        int f = sub + it * 16;
        float acc = 0.f;
#pragma unroll
        for (int j = 0; j < 16; ++j)
            acc += ajs[j] * z2[(size_t)sjs[j] * IN_D + f];
        float st = state[(size_t)node * IN_D + f];
        out_t[(size_t)node * 64 + f]      = st;     // state BEFORE update
        out_t[(size_t)node * 64 + 32 + f] = acc;    // xii
        state[(size_t)node * IN_D + f] = st + dt * acc;
    }
}

// ---------------------------------------------------------------------------
extern "C" void kernel_launch(void* const* d_in, const int* in_sizes, int n_in,
                              void* d_out, int out_size, void* d_ws, size_t ws_size,
                              hipStream_t stream) {
    const float* tarr  = (const float*)d_in[0];
    const float* x0    = (const float*)d_in[1];
    const int*   src   = (const int*)d_in[2];
    const int*   dst   = (const int*)d_in[3];
    const float* W_ih  = (const float*)d_in[4];
    const float* W_hh  = (const float*)d_in[5];
    const float* b_ih  = (const float*)d_in[6];
    const float* b_hh  = (const float*)d_in[7];
    const float* fc1   = (const float*)d_in[8];
    const float* attn1 = (const float*)d_in[9];
    const float* fc2   = (const float*)d_in[10];
    const float* attn2 = (const float*)d_in[11];
    float* out = (float*)d_out;

    char* ws = (char*)d_ws;
    size_t off = 0;
    auto alloc = [&](size_t nelem) {
        float* p = (float*)(ws + off);
        off += ((nelem * sizeof(float) + 255) / 256) * 256;
        return p;
    };
    float* state = alloc((size_t)N_NODES * IN_D);
    float* hA    = alloc((size_t)N_NODES * HID_D);
    float* hB    = alloc((size_t)N_NODES * HID_D);
    float* z1    = alloc((size_t)N_NODES * HID_D);
    float* z2    = alloc((size_t)N_NODES * IN_D);
    float* s1    = alloc(N_NODES);
    float* s2    = alloc(N_NODES);
    float* s1b   = alloc(N_NODES);
    float* s2b   = alloc(N_NODES);

    init_kernel<<<(N_NODES * HID_D + 255) / 256, 256, 0, stream>>>(x0, state, hA);

    float* hin  = hA;
    float* hout = hB;
    for (int t = 0; t < T_STEPS; ++t) {
        gru_z1_kernel<<<625, 256, 0, stream>>>(state, hin, hout, W_ih, W_hh, b_ih, b_hh,
                                               fc1, attn1, z1, s1, s2);
        agg1_z2_kernel<<<1250, 256, 0, stream>>>(z1, s1, s2, src, dst, fc2, attn2,
                                                 z2, s1b, s2b);
        agg2_final_kernel<<<1250, 256, 0, stream>>>(z2, s1b, s2b, src, dst, state,
                                                    out + (size_t)t * N_NODES * 64,
                                                    tarr, t);
        float* tmp = hin; hin = hout; hout = tmp;
    }
}